// BiLSTMCRF_31568009625987
// MI455X (gfx1250) — compile-verified
//
#include <hip/hip_runtime.h>
#include <hip/hip_bf16.h>
#include <stdint.h>

#define B_ 1024
#define L_ 512
#define K_ 48
#define PF_ 8   // prefetch distance in time steps

typedef __attribute__((ext_vector_type(2))) float v2f;
typedef __attribute__((ext_vector_type(8))) float v8f;

#define LN2_ 0.6931471805599453f

// Compiler-only fence: single-wave block, DS ops are in-order within a wave,
// so the sP store->load transpose needs no s_barrier / counter waits.
#define LDS_COMPILER_FENCE() asm volatile("" ::: "memory")

// Raw v_log_f32 (log2, no denorm fixup) -> natural log. Arguments here are
// always >= 0.018 (see call sites), so the fixup path __logf adds is dead.
__device__ __forceinline__ float fast_ln(float x) {
    return __builtin_amdgcn_logf(x) * LN2_;
}

// ---------------- Kernel 1: expT = exp(transition), 48x48 ----------------
__global__ void crf_exp_transition(const float* __restrict__ T,
                                   float* __restrict__ expT) {
    int i = blockIdx.x * blockDim.x + threadIdx.x;
    if (i < K_ * K_) expT[i] = __expf(T[i]);
}

// ---------------- Kernel 2: gold path score per batch element ----------------
__global__ void crf_path_score(const float* __restrict__ emit,
                               const float* __restrict__ T,
                               const int* __restrict__ tags,
                               const unsigned char* __restrict__ mask,
                               float* __restrict__ ts) {
    __shared__ float sT[K_ * K_];
    for (int i = threadIdx.x; i < K_ * K_; i += blockDim.x) sT[i] = T[i];
    __syncthreads();
    int b = blockIdx.x * blockDim.x + threadIdx.x;
    if (b >= B_) return;
    const float* eb = emit + (size_t)b * L_ * K_;
    const int* tg = tags + (size_t)b * L_;
    const unsigned char* mk = mask + (size_t)b * L_;
    int prev = tg[0];
    float s = eb[prev];              // t=0 always inside mask (lengths >= 2)
    for (int t = 1; t < L_; ++t) {   // mask is a prefix of ones
        if (!mk[t]) break;
        int cur = tg[t];
        s += eb[t * K_ + cur] + sT[prev * K_ + cur];
        prev = cur;
    }
    ts[b] = s;
}

// ---------------- Kernel 3: forward algorithm via f32 WMMA ----------------
// One wave (32 threads) per block handles 16 batch rows x 48 states.
// C/D layout: VGPR v, lane l -> row = v + 8*(l/16), col = 16n + (l&15).
// A layout:   lane l -> row = l&15, k = 2*(l/16) + g  (g = vgpr 0/1).
// B layout:   VGPR g, lane l -> i = 4kk + 2*(l/16) + g, j = 16n + (l&15).
__launch_bounds__(32)
__global__ void crf_forward(const float* __restrict__ emit,
                            const float* __restrict__ expT,
                            const unsigned char* __restrict__ mask,
                            const float* __restrict__ ts,
                            float* __restrict__ out) {
    __shared__ float sP[16 * 50];   // p staging, stride 50 to spread banks
    __shared__ int sLen[16];

    const int lane = threadIdx.x;
    const int half = lane >> 4;
    const int lc = lane & 15;
    const int b0 = blockIdx.x * 16;

    // ---- sequence lengths from the prefix mask (byte-sum trick) ----
    {
        const unsigned int* mrow =
            (const unsigned int*)(mask + (size_t)(b0 + lc) * L_);
        int len = 0;
        for (int i = 0; i < L_ / 4; ++i) {
            unsigned int u = mrow[i];                 // bytes are 0/1
            len += (int)((u * 0x01010101u) >> 24);    // sum of 4 bytes
        }
        if (half == 0) sLen[lc] = len;
    }
    __syncthreads();
    int flen[8];
#pragma unroll
    for (int v = 0; v < 8; ++v) flen[v] = sLen[v + 8 * half];
    const int maxlen = sLen[0];   // lengths sorted descending -> group max

    // ---- resident B fragments of exp(T): 3 n-tiles x 12 k-steps x 2 ----
    float bf[3][12][2];
#pragma unroll
    for (int n = 0; n < 3; ++n)
#pragma unroll
        for (int kk = 0; kk < 12; ++kk)
#pragma unroll
            for (int g = 0; g < 2; ++g) {
                int i = 4 * kk + 2 * half + g;
                int j = 16 * n + lc;
                bf[n][kk][g] = expT[i * K_ + j];
            }

    // ---- per-slot emit offsets (single base array; prefetch folds the
    // lane-column rotation into one per-lane delta to save VGPRs) ----
    int rowbase[8];
#pragma unroll
    for (int v = 0; v < 8; ++v)
        rowbase[v] = ((b0 + v + 8 * half) * L_) * K_ + lc;
    const int pfoff = (lc % 3) * 16 - lc;  // lanes cover all 3x64B row chunks

    // ---- d0 = emit[:, 0, :] ----
    float d[3][8];
#pragma unroll
    for (int v = 0; v < 8; ++v) {
        const float* pv = emit + rowbase[v];
        d[0][v] = pv[0]; d[1][v] = pv[16]; d[2][v] = pv[32];
    }

    // ---- software pipeline: preload emit for t=1 ----
    float e[3][8];
#pragma unroll
    for (int v = 0; v < 8; ++v) {
        const float* pv = emit + rowbase[v] + K_;
        e[0][v] = pv[0]; e[1][v] = pv[16]; e[2][v] = pv[32];
    }

    // ---- sequential recursion over time ----
    for (int t = 1; t < maxlen; ++t) {
        // prefetch step t+PF_ (covers cold-HBM latency; no counter traffic)
        if (t + PF_ < L_) {
            const int tpfK = (t + PF_) * K_ + pfoff;
#pragma unroll
            for (int v = 0; v < 8; ++v)
                __builtin_prefetch(emit + rowbase[v] + tpfK, 0, 3);
        }
        // issue next step's emit loads now; consumed next iteration
        float en[3][8];
        {
            const int tload = (t + 1 < maxlen) ? (t + 1) : t;  // stay in bounds
            const int tK = tload * K_;
#pragma unroll
            for (int v = 0; v < 8; ++v) {
                const float* pv = emit + rowbase[v] + tK;
                en[0][v] = pv[0]; en[1][v] = pv[16]; en[2][v] = pv[32];
            }
        }

        // row max over all 48 cols (3 tiles, then across the 16-lane half)
        float m[8];
#pragma unroll
        for (int v = 0; v < 8; ++v) {
            float mx = fmaxf(fmaxf(d[0][v], d[1][v]), d[2][v]);
#pragma unroll
            for (int o = 1; o < 16; o <<= 1)
                mx = fmaxf(mx, __shfl_xor(mx, o, 32));
            m[v] = mx;
        }

        // p = exp(d - m) staged to LDS (C/D layout -> A layout transpose)
#pragma unroll
        for (int n = 0; n < 3; ++n)
#pragma unroll
            for (int v = 0; v < 8; ++v)
                sP[(v + 8 * half) * 50 + 16 * n + lc] =
                    __expf(d[n][v] - m[v]);
        LDS_COMPILER_FENCE();

        // S = p x exp(T): 6 independent WMMA chains (2 partials x 3 n-tiles)
        v8f acc[2][3] = {{{}, {}, {}}, {{}, {}, {}}};
#pragma unroll
        for (int kk = 0; kk < 12; ++kk) {
            v2f a = *(const v2f*)&sP[lc * 50 + 4 * kk + 2 * half];
            v2f b0v = {bf[0][kk][0], bf[0][kk][1]};
            v2f b1v = {bf[1][kk][0], bf[1][kk][1]};
            v2f b2v = {bf[2][kk][0], bf[2][kk][1]};
            const int s = kk & 1;
            acc[s][0] = __builtin_amdgcn_wmma_f32_16x16x4_f32(
                false, a, false, b0v, (short)0, acc[s][0], false, false);
            acc[s][1] = __builtin_amdgcn_wmma_f32_16x16x4_f32(
                false, a, false, b1v, (short)0, acc[s][1], false, false);
            acc[s][2] = __builtin_amdgcn_wmma_f32_16x16x4_f32(
                false, a, false, b2v, (short)0, acc[s][2], false, false);
        }
        LDS_COMPILER_FENCE();   // sP reused next iteration (DS in-order)

        v8f S0 = acc[0][0] + acc[1][0];
        v8f S1 = acc[0][1] + acc[1][1];
        v8f S2 = acc[0][2] + acc[1][2];

        // d = mask ? emit + m + ln(S) : d   (cndmask keeps EXEC all-ones)
        // S >= min(exp(T)) ~ 0.018: raw v_log_f32 is safe (no denorm fixup).
#pragma unroll
        for (int v = 0; v < 8; ++v) {
            bool on = (t < flen[v]);
            float base = e[0][v];   // keep 3 independent FMA streams
            float dn0 = fmaf(__builtin_amdgcn_logf(S0[v]), LN2_,
                             e[0][v] + m[v]);
            float dn1 = fmaf(__builtin_amdgcn_logf(S1[v]), LN2_,
                             e[1][v] + m[v]);
            float dn2 = fmaf(__builtin_amdgcn_logf(S2[v]), LN2_,
                             e[2][v] + m[v]);
            (void)base;
            d[0][v] = on ? dn0 : d[0][v];
            d[1][v] = on ? dn1 : d[1][v];
            d[2][v] = on ? dn2 : d[2][v];
        }

        // rotate the emit pipeline (register renames, no real moves)
#pragma unroll
        for (int n = 0; n < 3; ++n)
#pragma unroll
            for (int v = 0; v < 8; ++v)
                e[n][v] = en[n][v];
    }

    // ---- log_z = logsumexp over 48 states, then final loss ----
    float logz[8];
#pragma unroll
    for (int v = 0; v < 8; ++v) {
        float mx = fmaxf(fmaxf(d[0][v], d[1][v]), d[2][v]);
#pragma unroll
        for (int o = 1; o < 16; o <<= 1)
            mx = fmaxf(mx, __shfl_xor(mx, o, 32));
        float s = __expf(d[0][v] - mx) + __expf(d[1][v] - mx) +
                  __expf(d[2][v] - mx);
#pragma unroll
        for (int o = 1; o < 16; o <<= 1) s += __shfl_xor(s, o, 32);
        // s >= 1 (the max element contributes exp(0)); raw v_log is safe
        logz[v] = fmaf(__builtin_amdgcn_logf(s), LN2_, mx);
    }
    if (lc == 0) {   // lanes 0 and 16 -> rows 0..7 and 8..15
#pragma unroll
        for (int v = 0; v < 8; ++v) {
            int row = b0 + v + 8 * half;
            out[row] = -(ts[row] - logz[v]);
        }
    }
}

// ---------------- launcher ----------------
extern "C" void kernel_launch(void* const* d_in, const int* in_sizes, int n_in,
                              void* d_out, int out_size, void* d_ws,
                              size_t ws_size, hipStream_t stream) {
    (void)in_sizes; (void)n_in; (void)out_size; (void)ws_size;
    const float* emit = (const float*)d_in[0];           // (B, L, K) f32
    const float* T = (const float*)d_in[1];              // (K, K) f32
    const int* tags = (const int*)d_in[2];               // (B, L) i32
    const unsigned char* mask = (const unsigned char*)d_in[3]; // (B, L) bool
    float* out = (float*)d_out;                          // (B,) f32

    float* expT = (float*)d_ws;          // 2304 floats
    float* ts = expT + K_ * K_;          // 1024 floats

    hipLaunchKernelGGL(crf_exp_transition, dim3((K_ * K_ + 255) / 256),
                       dim3(256), 0, stream, T, expT);
    hipLaunchKernelGGL(crf_path_score, dim3(B_ / 256), dim3(256), 0, stream,
                       emit, T, tags, mask, ts);
    hipLaunchKernelGGL(crf_forward, dim3(B_ / 16), dim3(32), 0, stream,
                       emit, expT, mask, ts, out);
}